// Scatter2D_80874234184357
// MI455X (gfx1250) — compile-verified
//
#include <hip/hip_runtime.h>

// ---------------- Problem constants (from reference setup_inputs) ----------
#define BATCH 16                 // B
#define NTOT  1179648            // N = 18 * 256 * 256
#define XMAX  2048
#define YMAX  1024
#define TPB   256                // 8 waves (wave32)

#define HIST_BLOCKS 256
#define CHUNK (NTOT / HIST_BLOCKS)   // 4608 n per block
#define TILE  256                    // n per staged tile (one n per thread)
#define TILES (CHUNK / TILE)         // 18 tiles per block
#define NBUF  3                      // triple-buffered TDM staging

typedef __attribute__((ext_vector_type(4))) unsigned int u32x4;
typedef __attribute__((ext_vector_type(8))) int          i32x8;
typedef __attribute__((ext_vector_type(4))) int          i32x4;

// Only the DEVICE pass may look for / use the TDM builtins. (This hipcc
// defines __AMDGCN__ even on the host pass, so use __HIP_DEVICE_COMPILE__.)
#if defined(__HIP_DEVICE_COMPILE__)
#if !__has_builtin(__builtin_amdgcn_tensor_load_to_lds)
#error "gfx1250 DEVICE pass lacks __builtin_amdgcn_tensor_load_to_lds"
#endif
#define USE_TDM 1
#else
#define USE_TDM 0
#endif

// ---------------------------------------------------------------------------
// TDM issue: DMA a [BATCH x TILE] f32 tile of x (row stride NTOT elements)
// into LDS at lds_off. D# built per CDNA5 ISA 8.3/8.4:
//   group0: count=1 | lds_addr | global_addr[56:0] | type=2
//   group1: data_size=4B, tensor_dim0=NTOT, tensor_dim1=BATCH,
//           tile_dim0=TILE, tile_dim1=BATCH, tensor_dim0_stride=NTOT
// ---------------------------------------------------------------------------
#if USE_TDM
__device__ __forceinline__ void tdm_issue(const float* gx, int base,
                                          unsigned lds_off) {
  unsigned long long ga = (unsigned long long)(const void*)(gx + base);
  u32x4 g0;
  g0[0] = 1u;                                    // count=1, user mode
  g0[1] = lds_off;                               // lds_addr (bytes)
  g0[2] = (unsigned)ga;                          // global_addr[31:0]
  g0[3] = (unsigned)((ga >> 32) & 0x1FFFFFFu)    // global_addr[56:32]
        | (2u << 30);                            // type=2 ("image")
  i32x8 g1 = (i32x8)0;
  g1[0] = 0x00020000;                            // data_size = 4 bytes
  g1[1] = (int)((NTOT & 0xFFFF) << 16);          // tensor_dim0[15:0] @ bit48
  g1[2] = (int)(((NTOT >> 16) & 0xFFFF)          // tensor_dim0[31:16]
        | (BATCH << 16));                        // tensor_dim1[15:0] @ bit80
  g1[3] = (int)(TILE << 16);                     // tile_dim0 @ bit112
  g1[4] = BATCH;                                 // tile_dim1 @ bit128
  g1[5] = NTOT;                                  // tensor_dim0_stride[31:0]
  i32x4 g2 = (i32x4)0;
  i32x4 g3 = (i32x4)0;
#if __clang_major__ >= 23
  __builtin_amdgcn_tensor_load_to_lds(g0, g1, g2, g3, (i32x8)0, 0);
#else
  __builtin_amdgcn_tensor_load_to_lds(g0, g1, g2, g3, 0);
#endif
}
#endif  // USE_TDM

__device__ __forceinline__ void stage_tile(const float* __restrict__ x,
                                           int base, float* buf) {
#if USE_TDM
  if (threadIdx.x < 32) {                        // wave 0 only: 1 DMA per tile
    unsigned lds_off = (unsigned)(unsigned long long)buf;  // low 32b = LDS off
    tdm_issue(x, base, lds_off);
  }
#else
  (void)x; (void)base; (void)buf;
#endif
}

// Wait until all but `pending` TDM ops (wave 0) have retired, then barrier.
__device__ __forceinline__ void wait_tile(int pending) {
#if USE_TDM
  if (threadIdx.x < 32) {
    switch (pending) {
      case 0:  __builtin_amdgcn_s_wait_tensorcnt(0); break;
      case 1:  __builtin_amdgcn_s_wait_tensorcnt(1); break;
      default: __builtin_amdgcn_s_wait_tensorcnt(2); break;
    }
  }
#else
  (void)pending;
#endif
  __syncthreads();
}

// ---------------------------------------------------------------------------
// K1: zero the global accumulator (sum[16*2048] + cnt[2048]) in d_ws
// ---------------------------------------------------------------------------
__global__ void __launch_bounds__(TPB) zero_acc_kernel(float* __restrict__ acc) {
  int i = blockIdx.x * TPB + threadIdx.x;
  if (i < BATCH * XMAX + XMAX) acc[i] = 0.0f;
}

// ---------------------------------------------------------------------------
// K2: TDM-staged, LDS-privatized histogram.
//   LDS: hist[16][2048] (b-major so random cols spread over all 64 banks),
//        cnt[2048], triple-buffered stage[3][16][256].  Total 188,416 B
//   (fits the 320 KB WGP budget; TDM keeps 2 tile-DMAs in flight).
// ---------------------------------------------------------------------------
__global__ void __launch_bounds__(TPB)
hist_kernel(const float* __restrict__ x, const int* __restrict__ xc,
            float* __restrict__ acc_sum, float* __restrict__ acc_cnt) {
  __shared__ float s_hist[BATCH * XMAX];         // 131072 B
  __shared__ float s_cnt[XMAX];                  //   8192 B
  __shared__ float s_stage[NBUF][BATCH * TILE];  //  49152 B

  const int tid = threadIdx.x;
  for (int i = tid; i < BATCH * XMAX; i += TPB) s_hist[i] = 0.0f;
  for (int i = tid; i < XMAX; i += TPB) s_cnt[i] = 0.0f;
  __syncthreads();

  const int chunk0 = blockIdx.x * CHUNK;

  // Prologue: start two DMAs.
  stage_tile(x, chunk0 + 0 * TILE, &s_stage[0][0]);
  stage_tile(x, chunk0 + 1 * TILE, &s_stage[1][0]);

  for (int t = 0; t < TILES; ++t) {
    const int buf = t % NBUF;
    if (t + 2 < TILES)                           // keep pipeline depth 2
      stage_tile(x, chunk0 + (t + 2) * TILE, &s_stage[(t + 2) % NBUF][0]);

    // column index for this thread's element (plain coalesced global load,
    // overlaps with the in-flight DMAs)
    const int col = xc[chunk0 + t * TILE + tid];

    const int rem = TILES - 1 - t;               // newer tiles outstanding
    wait_tile(rem > 2 ? 2 : rem);                // tile t retired + barrier

    atomicAdd(&s_cnt[col], 1.0f);                // ds_add_f32
    const float* xs = &s_stage[buf][0];
#pragma unroll
    for (int b = 0; b < BATCH; ++b)
      atomicAdd(&s_hist[b * XMAX + col], xs[b * TILE + tid]);

    __syncthreads();                             // buf consumed -> reusable
  }

  // merge block-private histogram into the global accumulator (coalesced
  // global_atomic_add_f32; 128 + 8 atomics per thread)
  for (int i = tid; i < BATCH * XMAX; i += TPB) atomicAdd(&acc_sum[i], s_hist[i]);
  for (int i = tid; i < XMAX; i += TPB)         atomicAdd(&acc_cnt[i], s_cnt[i]);
}

// ---------------------------------------------------------------------------
// K3: colval[b][j] = acc_sum[b][j] / max(cnt[j], 1)
// ---------------------------------------------------------------------------
__global__ void __launch_bounds__(TPB)
colval_kernel(const float* __restrict__ acc_sum,
              const float* __restrict__ acc_cnt, float* __restrict__ colval) {
  int i = blockIdx.x * TPB + threadIdx.x;
  if (i >= BATCH * XMAX) return;
  int j = i & (XMAX - 1);
  float c = acc_cnt[j];
  c = (c < 1.0f) ? 1.0f : c;
  colval[i] = acc_sum[i] / c;
}

// ---------------------------------------------------------------------------
// K4: fused zero + sparse scatter of the 128 MB output:
//   out[b][y][j] = (y == y_coord[j]) ? colval[b][j] : 0
// float4 per thread, fully coalesced stores; y_coord/colval live in L2.
// Index math is pure shifts/masks: J4=512, YMAX=1024.
// ---------------------------------------------------------------------------
__global__ void __launch_bounds__(TPB)
write_out_kernel(const float* __restrict__ colval, const int* __restrict__ yc,
                 float* __restrict__ out) {
  const unsigned total = (unsigned)BATCH * YMAX * (XMAX / 4);  // 8,388,608
  unsigned i = blockIdx.x * TPB + threadIdx.x;
  if (i >= total) return;
  unsigned j0 = (i & 511u) << 2;        // column group
  unsigned y  = (i >> 9) & 1023u;       // output row
  unsigned b  = i >> 19;                // batch

  const int4   yv = *(const int4*)(yc + j0);
  const float4 cv = *(const float4*)(colval + (b << 11) + j0);
  float4 o;
  o.x = (yv.x == (int)y) ? cv.x : 0.0f;
  o.y = (yv.y == (int)y) ? cv.y : 0.0f;
  o.z = (yv.z == (int)y) ? cv.z : 0.0f;
  o.w = (yv.w == (int)y) ? cv.w : 0.0f;
  *(float4*)(out + (((size_t)b << 10) + y) * XMAX + j0) = o;
}

// ---------------------------------------------------------------------------
extern "C" void kernel_launch(void* const* d_in, const int* in_sizes, int n_in,
                              void* d_out, int out_size, void* d_ws,
                              size_t ws_size, hipStream_t stream) {
  const float* x    = (const float*)d_in[0];   // [16,1,1,N] f32
  const int*   xc   = (const int*)d_in[1];     // [N] int32
  const int*   ycrd = (const int*)d_in[2];     // [2048] int32
  // d_in[3]=x_max, d_in[4]=y_max (compile-time constants here)

  float* ws      = (float*)d_ws;
  float* acc_sum = ws;                               // 32768 f32
  float* acc_cnt = ws + BATCH * XMAX;                //  2048 f32
  float* colval  = ws + BATCH * XMAX + XMAX;         // 32768 f32
  float* out     = (float*)d_out;                    // [16,1024,2048] f32

  const int accN = BATCH * XMAX + XMAX;
  zero_acc_kernel<<<(accN + TPB - 1) / TPB, TPB, 0, stream>>>(acc_sum);

  hist_kernel<<<HIST_BLOCKS, TPB, 0, stream>>>(x, xc, acc_sum, acc_cnt);

  colval_kernel<<<(BATCH * XMAX + TPB - 1) / TPB, TPB, 0, stream>>>(
      acc_sum, acc_cnt, colval);

  const unsigned total4 = (unsigned)BATCH * YMAX * (XMAX / 4);
  write_out_kernel<<<(total4 + TPB - 1) / TPB, TPB, 0, stream>>>(
      colval, ycrd, out);
}